// GCNLayer_352187319192
// MI455X (gfx1250) — compile-verified
//
#include <hip/hip_runtime.h>

// GCN layer for MI455X (gfx1250, wave32).
// Dominant cost: edge gather/scatter (~1.7 GB -> ~75-150 us @ 23.3 TB/s HBM).
// GEMM h = x@W^T + b done with v_wmma_f32_16x16x4_f32 (fp32 precision kept).

#define IN_CH  128
#define OUT_CH 128

typedef float v2f __attribute__((ext_vector_type(2)));
typedef float v8f __attribute__((ext_vector_type(8)));

// ---------------- degree kernels ----------------
__global__ void k_deg_init(float* __restrict__ deg, int n) {
    int i = blockIdx.x * blockDim.x + threadIdx.x;
    if (i < n) deg[i] = 1.0f;                      // self-loop contribution
}

__global__ void k_deg_count(const long long* __restrict__ dst, float* __restrict__ deg, int e) {
    int i = blockIdx.x * blockDim.x + threadIdx.x;
    if (i < e) atomicAdd(&deg[(int)dst[i]], 1.0f);
}

__global__ void k_deg_rsqrt(float* __restrict__ deg, int n) {
    int i = blockIdx.x * blockDim.x + threadIdx.x;
    if (i < n) deg[i] = rsqrtf(deg[i]);            // deg -> dinv in place (deg >= 1, no inf)
}

// ---------------- WMMA GEMM: h = x @ W^T + b ----------------
// One block = 256 threads = 8 waves; each wave computes a 16(M) x 128(N) strip.
// W staged in LDS, pair-packed over K so B fragments are 8-byte ds loads:
//   ldsW[ ((k>>1)*OUT_CH + n)*2 + (k&1) ] = W[n][k]
__global__ __launch_bounds__(256) void k_gemm_wmma(const float* __restrict__ x,
                                                   const float* __restrict__ W,
                                                   const float* __restrict__ bias,
                                                   float* __restrict__ h, int n) {
    __shared__ float ldsW[IN_CH * OUT_CH];         // 64 KB (WGP has 320 KB)

    int t = threadIdx.x;
    for (int idx = t; idx < IN_CH * OUT_CH; idx += 256) {
        int nr = idx >> 7;       // row of W (output channel)
        int k  = idx & 127;      // col of W (input channel)
        ldsW[(((k >> 1) * OUT_CH + nr) << 1) | (k & 1)] = W[idx];
    }
    __syncthreads();

    int lane = t & 31;
    int wave = t >> 5;
    int hi   = lane >> 4;        // 0: lanes 0-15, 1: lanes 16-31
    int l15  = lane & 15;

    int m0 = (blockIdx.x * 8 + wave) * 16;         // first row of this wave's tile

    v8f acc[8] = {};                               // 8 N-tiles of 16x16 f32 C

    // A fragment addressing (16x4 f32): lane l15 = row M, hi selects K pair,
    // vector element selects K%2 -> a = x[m, k0 + 2*hi + {0,1}]
    int mrow = m0 + l15;
    if (mrow >= n) mrow = n - 1;                   // clamp tail rows (stores predicated)
    const float* xrow = x + (size_t)mrow * IN_CH + 2 * hi;

    for (int k0 = 0; k0 < IN_CH; k0 += 4) {
        v2f a = *(const v2f*)(xrow + k0);          // 8B-aligned global load
        int kbase = (((k0 >> 1) + hi) * OUT_CH) << 1;
#pragma unroll
        for (int nt = 0; nt < 8; ++nt) {
            int bidx = kbase + ((nt * 16 + l15) << 1);
            v2f bf = { ldsW[bidx], ldsW[bidx + 1] };   // B(k, n) = W[n][k]
            // 8 args: (neg_a, A, neg_b, B, c_mod, C, reuse_a, reuse_b)
            acc[nt] = __builtin_amdgcn_wmma_f32_16x16x4_f32(
                false, a, false, bf, (short)0, acc[nt], false, false);
        }
    }

    // C layout: vgpr r, half -> element (M = r + 8*hi, N = l15)
#pragma unroll
    for (int nt = 0; nt < 8; ++nt) {
        int ncol = nt * 16 + l15;
        float bv = bias[ncol];
#pragma unroll
        for (int r = 0; r < 8; ++r) {
            int row = m0 + r + 8 * hi;
            if (row < n) h[(size_t)row * OUT_CH + ncol] = acc[nt][r] + bv;
        }
    }
}

// ---------------- self-loop term: out[i] = h[i] * dinv[i]^2 ----------------
__global__ void k_self(const float* __restrict__ h, const float* __restrict__ dinv,
                       float* __restrict__ out, int total) {
    int gid = blockIdx.x * blockDim.x + threadIdx.x;
    if (gid < total) {
        int i = gid >> 7;
        float di = dinv[i];
        out[gid] = h[gid] * (di * di);
    }
}

// ---------------- edge scatter: out[dst] += h[src] * dinv[src]*dinv[dst] ----------------
// One wave32 per edge; each lane handles 4 channels (float4 gather, 4 f32 atomics).
__global__ __launch_bounds__(256) void k_scatter(const float* __restrict__ h,
                                                 const long long* __restrict__ src,
                                                 const long long* __restrict__ dst,
                                                 const float* __restrict__ dinv,
                                                 float* __restrict__ out, int e) {
    int gid  = blockIdx.x * blockDim.x + threadIdx.x;
    int edge = gid >> 5;
    int lane = gid & 31;
    if (edge >= e) return;
    int s = (int)src[edge];
    int d = (int)dst[edge];
    float w = dinv[s] * dinv[d];
    float4 v = *(const float4*)(h + (size_t)s * OUT_CH + lane * 4);   // coalesced 512B/wave
    float* op = out + (size_t)d * OUT_CH + lane * 4;
    atomicAdd(op + 0, v.x * w);
    atomicAdd(op + 1, v.y * w);
    atomicAdd(op + 2, v.z * w);
    atomicAdd(op + 3, v.w * w);
}

extern "C" void kernel_launch(void* const* d_in, const int* in_sizes, int n_in,
                              void* d_out, int out_size, void* d_ws, size_t ws_size,
                              hipStream_t stream) {
    const float*     x  = (const float*)d_in[0];
    const long long* ei = (const long long*)d_in[1];   // int64 edge_index [2, E]
    const float*     W  = (const float*)d_in[2];
    const float*     b  = (const float*)d_in[3];
    int n = in_sizes[0] / IN_CH;
    int e = in_sizes[1] / 2;
    const long long* srcp = ei;
    const long long* dstp = ei + e;
    float* out = (float*)d_out;

    // workspace: h [n*128 floats] then deg/dinv [n floats]
    float* h   = (float*)d_ws;
    float* deg = h + (size_t)n * OUT_CH;

    k_deg_init <<<(n + 255) / 256, 256, 0, stream>>>(deg, n);
    k_deg_count<<<(e + 255) / 256, 256, 0, stream>>>(dstp, deg, e);
    k_deg_rsqrt<<<(n + 255) / 256, 256, 0, stream>>>(deg, n);

    int mblocks = (n + 127) / 128;                 // 128 rows per block (8 waves x 16)
    k_gemm_wmma<<<mblocks, 256, 0, stream>>>(x, W, b, h, n);

    int total = n * OUT_CH;
    k_self<<<(total + 255) / 256, 256, 0, stream>>>(h, deg, out, total);

    long long sthreads = (long long)e * 32;        // one wave per edge
    k_scatter<<<(int)((sthreads + 255) / 256), 256, 0, stream>>>(h, srcp, dstp, deg, out, e);
}